// ResidualAttentionBlockTensorCache_16973710754308
// MI455X (gfx1250) — compile-verified
//
#include <hip/hip_runtime.h>
#include <math.h>

// ---------------- problem constants ----------------
#define BATCH   16
#define CDIM    1280
#define NHEAD   20
#define DHEAD   64
#define CACHE_T 448
#define TCROSS  1500
#define MLPDIM  5120
#define LN_EPS  1e-5f

typedef __attribute__((ext_vector_type(2))) float v2f;
typedef __attribute__((ext_vector_type(4))) float v4f;
typedef __attribute__((ext_vector_type(8))) float v8f;

// =====================================================================
// LayerNorm: one block per row of [BATCH, CDIM]
// =====================================================================
__global__ __launch_bounds__(256)
void ln_kernel(const float* __restrict__ x, const float* __restrict__ g,
               const float* __restrict__ b, float* __restrict__ out) {
    __shared__ float red[256];
    const int row = blockIdx.x;
    const int tid = threadIdx.x;
    const float* xr = x + (size_t)row * CDIM;

    float s = 0.f, ss = 0.f;
    for (int i = tid; i < CDIM; i += 256) { float v = xr[i]; s += v; ss += v * v; }
    red[tid] = s; __syncthreads();
    for (int o = 128; o > 0; o >>= 1) { if (tid < o) red[tid] += red[tid + o]; __syncthreads(); }
    const float mean = red[0] / (float)CDIM; __syncthreads();
    red[tid] = ss; __syncthreads();
    for (int o = 128; o > 0; o >>= 1) { if (tid < o) red[tid] += red[tid + o]; __syncthreads(); }
    const float var = red[0] / (float)CDIM - mean * mean;
    const float rstd = rsqrtf(var + LN_EPS);
    for (int i = tid; i < CDIM; i += 256)
        out[(size_t)row * CDIM + i] = (xr[i] - mean) * rstd * g[i] + b[i];
}

// =====================================================================
// [16,K] x [K,N] GEMM with V_WMMA_F32_16X16X4_F32, K split across 8 waves.
// Block = 256 threads = 8 waves; each wave computes a partial 16x16 C tile
// over its K/8 chunk, then 8-way LDS reduction. EXEC uniform through WMMAs.
// Weight stream: non-temporal loads + global_prefetch ahead.
// flags bit0 = exact GELU on output.
// =====================================================================
__global__ __launch_bounds__(256)
void gemm16_wmma(const float* __restrict__ A, const float* __restrict__ W,
                 const float* __restrict__ bias, const float* __restrict__ resid,
                 float* __restrict__ out, int K, int N, int flags) {
    __shared__ float part[8][256];
    const int tid  = threadIdx.x;
    const int wave = tid >> 5;
    const int lane = tid & 31;
    const int col  = lane & 15;
    const int kp   = lane >> 4;          // 0/1 -> K-pair select within 16x4 tile
    const int n0   = blockIdx.x << 4;

    const int kchunk = K >> 3;           // K/8 (1280->160, 5120->640); mult of 16
    const int kbeg = wave * kchunk;
    const int kend = kbeg + kchunk;

    v8f acc = {};
    const float* Ap = A + (size_t)col * K + 2 * kp;
    const float* Wp = W + (size_t)(2 * kp) * N + n0 + col;

    for (int k0 = kbeg; k0 < kend; k0 += 16) {
        int kpre = k0 + 64; if (kpre > K - 4) kpre = K - 4;
        __builtin_prefetch(&Wp[(size_t)kpre * N], 0, 1);
        #pragma unroll
        for (int kk = 0; kk < 16; kk += 4) {
            const int ka = k0 + kk;
            v2f a;
            a.x = Ap[ka];
            a.y = Ap[ka + 1];
            v2f bm;
            bm.x = __builtin_nontemporal_load(&Wp[(size_t)ka * N]);
            bm.y = __builtin_nontemporal_load(&Wp[(size_t)ka * N + N]);
            acc = __builtin_amdgcn_wmma_f32_16x16x4_f32(
                      false, a, false, bm, (short)0, acc, false, false);
        }
    }

    #pragma unroll
    for (int r = 0; r < 8; ++r) part[wave][lane + 32 * r] = acc[r];
    __syncthreads();

    // thread t owns output element (lane=t&31, r=t>>5); part index == t
    const int l2 = tid & 31;
    const int r2 = tid >> 5;
    const int c2 = l2 & 15;
    const int m2 = r2 + 8 * (l2 >> 4);
    float v = 0.f;
    #pragma unroll
    for (int w = 0; w < 8; ++w) v += part[w][tid];
    if (bias) v += bias[n0 + c2];
    if (flags & 1) v = 0.5f * v * (1.0f + erff(v * 0.70710678118654752f)); // exact GELU
    if (resid) v += resid[(size_t)m2 * N + n0 + c2];
    out[(size_t)m2 * N + n0 + c2] = v;
}

// =====================================================================
// KV-cache shift: out[:, :CACHE_T-1, :] = old[:, 1:, :]; out[:, -1, :] = new
// 128-bit granularity (CDIM/4 = 320 vectors per row), non-temporal stream
// =====================================================================
__global__ __launch_bounds__(256)
void shift_cache_kernel(const float* __restrict__ oldc, const float* __restrict__ newrow,
                        float* __restrict__ outc) {
    const int C4 = CDIM / 4;
    const size_t total = (size_t)BATCH * CACHE_T * C4;
    size_t idx = (size_t)blockIdx.x * blockDim.x + threadIdx.x;
    if (idx >= total) return;
    const int c = idx % C4;
    const int t = (idx / C4) % CACHE_T;
    const int b = idx / ((size_t)C4 * CACHE_T);
    const v4f* o4 = (const v4f*)oldc;
    const v4f* n4 = (const v4f*)newrow;
    v4f* out4 = (v4f*)outc;
    out4[idx] = (t < CACHE_T - 1)
                  ? __builtin_nontemporal_load(&o4[((size_t)b * CACHE_T + t + 1) * C4 + c])
                  : n4[(size_t)b * C4 + c];
}

// =====================================================================
// Single-query attention for one (batch, head): scores + softmax + AV.
// scale 0.125 = dhead^-0.5 (Whisper applies dhead^-0.25 to q and k each)
// =====================================================================
__global__ __launch_bounds__(256)
void attn1_kernel(const float* __restrict__ q, const float* __restrict__ kc,
                  const float* __restrict__ vc, const float* __restrict__ mask,
                  float* __restrict__ out, int T) {
    __shared__ float s[TCROSS];
    __shared__ float qs[DHEAD];
    __shared__ float red[256];
    const int tid = threadIdx.x;
    const int b = blockIdx.x / NHEAD;
    const int h = blockIdx.x % NHEAD;
    const size_t hoff = (size_t)h * DHEAD;

    for (int i = tid; i < DHEAD; i += 256) qs[i] = q[(size_t)b * CDIM + hoff + i];
    __syncthreads();

    float lmax = -3.4e38f;
    for (int j = tid; j < T; j += 256) {
        const float* kp = kc + ((size_t)b * T + j) * CDIM + hoff;
        float d = 0.f;
        #pragma unroll
        for (int i = 0; i < DHEAD; i += 4) {
            v4f kv = __builtin_nontemporal_load((const v4f*)(kp + i));
            d += kv.x * qs[i] + kv.y * qs[i + 1] + kv.z * qs[i + 2] + kv.w * qs[i + 3];
        }
        d = d * 0.125f + (mask ? mask[j] : 0.f);
        s[j] = d;
        lmax = fmaxf(lmax, d);
    }
    red[tid] = lmax; __syncthreads();
    for (int o = 128; o > 0; o >>= 1) { if (tid < o) red[tid] = fmaxf(red[tid], red[tid + o]); __syncthreads(); }
    const float mx = red[0]; __syncthreads();

    float lsum = 0.f;
    for (int j = tid; j < T; j += 256) { float e = expf(s[j] - mx); s[j] = e; lsum += e; }
    red[tid] = lsum; __syncthreads();
    for (int o = 128; o > 0; o >>= 1) { if (tid < o) red[tid] += red[tid + o]; __syncthreads(); }
    const float inv = 1.0f / red[0]; __syncthreads();

    // AV: 4 groups of 64 lanes split the T range, then cross-group reduce
    const int grp = tid >> 6;
    const int d0  = tid & 63;
    float acc = 0.f;
    for (int j = grp; j < T; j += 4) {
        const float* vp = vc + ((size_t)b * T + j) * CDIM + hoff + d0;
        acc += s[j] * __builtin_nontemporal_load(vp);
    }
    red[tid] = acc; __syncthreads();
    if (tid < 64)
        out[(size_t)b * CDIM + hoff + d0] =
            (red[d0] + red[64 + d0] + red[128 + d0] + red[192 + d0]) * inv;
}

// =====================================================================
// Host-side orchestration (graph-capture safe: launches only)
// =====================================================================
extern "C" void kernel_launch(void* const* d_in, const int* in_sizes, int n_in,
                              void* d_out, int out_size, void* d_ws, size_t ws_size,
                              hipStream_t stream) {
    const float* x       = (const float*)d_in[0];
    const float* skc     = (const float*)d_in[1];
    const float* svc     = (const float*)d_in[2];
    const float* cross_k = (const float*)d_in[3];
    const float* cross_v = (const float*)d_in[4];
    const float* mask    = (const float*)d_in[5];
    const float* ln1_g = (const float*)d_in[6],  *ln1_b = (const float*)d_in[7];
    const float* Wq = (const float*)d_in[8],  *bq = (const float*)d_in[9];
    const float* Wk = (const float*)d_in[10];
    const float* Wv = (const float*)d_in[11], *bv = (const float*)d_in[12];
    const float* Wo = (const float*)d_in[13], *bo = (const float*)d_in[14];
    const float* ln2_g = (const float*)d_in[15], *ln2_b = (const float*)d_in[16];
    const float* Wqc = (const float*)d_in[17], *bqc = (const float*)d_in[18];
    const float* Woc = (const float*)d_in[19], *boc = (const float*)d_in[20];
    const float* ln3_g = (const float*)d_in[21], *ln3_b = (const float*)d_in[22];
    const float* W1 = (const float*)d_in[23], *b1 = (const float*)d_in[24];
    const float* W2 = (const float*)d_in[25], *b2 = (const float*)d_in[26];

    // output regions: x [B,1,C], k_cache [B,CACHE,C], v_cache [B,CACHE,C]
    float* out_x = (float*)d_out;
    float* out_k = out_x + (size_t)BATCH * CDIM;
    float* out_v = out_k + (size_t)BATCH * CACHE_T * CDIM;

    // workspace layout (floats)
    float* ws  = (float*)d_ws;
    float* xn  = ws;                            // [16,1280]
    float* qb  = xn  + BATCH * CDIM;            // [16,1280]
    float* kb  = qb  + BATCH * CDIM;            // [16,1280]
    float* vb  = kb  + BATCH * CDIM;            // [16,1280]
    float* att = vb  + BATCH * CDIM;            // [16,1280]
    float* x1  = att + BATCH * CDIM;            // [16,1280]
    float* hb  = x1  + BATCH * CDIM;            // [16,5120]

    const dim3 blk256(256);
    const int gN  = CDIM / 16;                  // 80 tiles
    const int gM  = MLPDIM / 16;                // 320 tiles
    const int gSh = (int)(((size_t)BATCH * CACHE_T * (CDIM / 4) + 255) / 256);

    // ---- self-attention ----
    ln_kernel<<<BATCH, blk256, 0, stream>>>(x, ln1_g, ln1_b, xn);
    gemm16_wmma<<<gN, blk256, 0, stream>>>(xn, Wq, bq, nullptr, qb, CDIM, CDIM, 0);
    gemm16_wmma<<<gN, blk256, 0, stream>>>(xn, Wk, nullptr, nullptr, kb, CDIM, CDIM, 0);
    gemm16_wmma<<<gN, blk256, 0, stream>>>(xn, Wv, bv, nullptr, vb, CDIM, CDIM, 0);
    shift_cache_kernel<<<gSh, blk256, 0, stream>>>(skc, kb, out_k);
    shift_cache_kernel<<<gSh, blk256, 0, stream>>>(svc, vb, out_v);
    attn1_kernel<<<BATCH * NHEAD, blk256, 0, stream>>>(qb, out_k, out_v, mask, att, CACHE_T);
    gemm16_wmma<<<gN, blk256, 0, stream>>>(att, Wo, bo, x, x1, CDIM, CDIM, 0);

    // ---- cross-attention ----
    ln_kernel<<<BATCH, blk256, 0, stream>>>(x1, ln2_g, ln2_b, xn);
    gemm16_wmma<<<gN, blk256, 0, stream>>>(xn, Wqc, bqc, nullptr, qb, CDIM, CDIM, 0);
    attn1_kernel<<<BATCH * NHEAD, blk256, 0, stream>>>(qb, cross_k, cross_v, nullptr, att, TCROSS);
    gemm16_wmma<<<gN, blk256, 0, stream>>>(att, Woc, boc, x1, out_x, CDIM, CDIM, 0);

    // ---- MLP ----
    ln_kernel<<<BATCH, blk256, 0, stream>>>(out_x, ln3_g, ln3_b, xn);
    gemm16_wmma<<<gM, blk256, 0, stream>>>(xn, W1, b1, nullptr, hb, CDIM, MLPDIM, 1);
    gemm16_wmma<<<gN, blk256, 0, stream>>>(hb, W2, b2, out_x, out_x, MLPDIM, CDIM, 0);
}